// IntegralRayCasting_88192858456307
// MI455X (gfx1250) — compile-verified
//
#include <hip/hip_runtime.h>
#include <math.h>

// Problem constants (match reference)
#define HH 256
#define WW 256
#define BB 2
#define NN 256
#define ALPHA0 0.025f
#define BETAC 2.0f
#define EPSF 1e-8f
#define PIF 3.14159265358979323846f

typedef __attribute__((ext_vector_type(2))) float v2f;
typedef __attribute__((ext_vector_type(8))) float v8f;

// fast TRANS-pipe helpers (single v_rcp_f32 / v_rsq_f32)
__device__ __forceinline__ float frcp(float x) { return __builtin_amdgcn_rcpf(x); }
__device__ __forceinline__ float frsq(float x) { return __builtin_amdgcn_rsqf(x); }

// ---- sortable-uint encoding for float atomic max (handles negatives) ----
__device__ __forceinline__ unsigned enc_f(float f) {
  unsigned b = __float_as_uint(f);
  return (b & 0x80000000u) ? ~b : (b | 0x80000000u);
}
__device__ __forceinline__ float dec_f(unsigned u) {
  return (u & 0x80000000u) ? __uint_as_float(u ^ 0x80000000u)
                           : __uint_as_float(~u);
}

// ===========================================================================
// Kernel 1: per-batch scalars + per-Gaussian feature rows (16 floats each)
//   row = [a00,a11,a22,2a01,2a02,2a12, 0,0,  p0,p1,p2,0,  appR,appG,appB, msm]
//   A = rho * diag(lam_inv) * rho^T ; p = A mu ; msm = mu^T A mu
// ===========================================================================
__global__ void IRC_prep_kernel(const float* __restrict__ mu,
                                const float* __restrict__ rho,
                                const float* __restrict__ lam,
                                const float* __restrict__ app,
                                float* __restrict__ gfeat,
                                float* __restrict__ scal) {
  const int b = blockIdx.x;
  const int n = threadIdx.x;
  __shared__ float red[256];
  __shared__ float sScale;

  const int base = (b * NN + n) * 3;
  const float l0 = lam[base + 0], l1 = lam[base + 1], l2 = lam[base + 2];
  red[n] = fminf(l0, fminf(l1, l2));
  __syncthreads();
  for (int s = 128; s > 0; s >>= 1) {
    if (n < s) red[n] = fminf(red[n], red[n + s]);
    __syncthreads();
  }
  if (n == 0) {
    // scale = 10^ceil(log10(max(min lam, 1e-6)))
    float logs = ceilf(log10f(fmaxf(red[0], 1e-6f)));
    sScale = powf(10.0f, logs);
  }
  __syncthreads();
  const float scale = sScale;

  const float li0 = scale / fmaxf(l0, EPSF);
  const float li1 = scale / fmaxf(l1, EPSF);
  const float li2 = scale / fmaxf(l2, EPSF);
  const float li[3] = {li0, li1, li2};

  float R[3][3];
#pragma unroll
  for (int c = 0; c < 3; ++c)
#pragma unroll
    for (int k = 0; k < 3; ++k) R[c][k] = rho[base * 3 + c * 3 + k];

  float A[3][3];
#pragma unroll
  for (int c = 0; c < 3; ++c)
#pragma unroll
    for (int c2 = 0; c2 < 3; ++c2) {
      float s = 0.f;
#pragma unroll
      for (int k = 0; k < 3; ++k) s += R[c][k] * li[k] * R[c2][k];
      A[c][c2] = s;
    }

  const float m0 = mu[base + 0], m1 = mu[base + 1], m2 = mu[base + 2];
  const float p0 = A[0][0] * m0 + A[0][1] * m1 + A[0][2] * m2;
  const float p1 = A[1][0] * m0 + A[1][1] * m1 + A[1][2] * m2;
  const float p2 = A[2][0] * m0 + A[2][1] * m1 + A[2][2] * m2;
  const float msm = m0 * p0 + m1 * p1 + m2 * p2;

  float* g = gfeat + (b * NN + n) * 16;
  g[0] = A[0][0]; g[1] = A[1][1]; g[2] = A[2][2];
  g[3] = 2.0f * A[0][1]; g[4] = 2.0f * A[0][2]; g[5] = 2.0f * A[1][2];
  g[6] = 0.0f; g[7] = 0.0f;
  g[8] = p0; g[9] = p1; g[10] = p2; g[11] = 0.0f;
  g[12] = app[base + 0]; g[13] = app[base + 1]; g[14] = app[base + 2];
  g[15] = msm;

  if (n == 0) {
    const float alpha = ALPHA0 * scale;
    scal[b * 4 + 0] = alpha;
    scal[b * 4 + 1] = 0.5f * sqrtf(PIF * alpha);      // a coefficient
    ((unsigned*)scal)[b * 4 + 2] = enc_f(-3.0e38f);   // z-max slot init
  }
}

// ===========================================================================
// Kernel 2: per-batch max of z = rsm / max(rsr, eps) over all pixels/Gaussians
// ===========================================================================
__global__ void IRC_zmax_kernel(const float* __restrict__ gfeat,
                                const float* __restrict__ Kmat,
                                float* __restrict__ scal) {
  const int b = blockIdx.y;
  const int tid = threadIdx.x;
  __shared__ float sG[NN * 16];
  __shared__ float red[256];
  {
    float4* d = (float4*)sG;
    const float4* s = (const float4*)(gfeat + b * NN * 16);
    for (int i = tid; i < NN * 4; i += 256) d[i] = s[i];
  }
  __syncthreads();

  const int P = blockIdx.x * 256 + tid;
  const int x = P & (WW - 1);
  const int y = P >> 8;
  const float fx = Kmat[b * 9 + 0], fy = Kmat[b * 9 + 4];
  const float cx = Kmat[b * 9 + 2], cy = Kmat[b * 9 + 5];
  float rx = ((float)x - cx) / fx;
  float ry = ((float)y - cy) / fy;
  float rz = 1.0f;
  const float inv = rsqrtf(rx * rx + ry * ry + 1.0f);
  rx *= inv; ry *= inv; rz *= inv;
  const float q0 = rx * rx, q1 = ry * ry, q2 = rz * rz;
  const float q3 = rx * ry, q4 = rx * rz, q5 = ry * rz;

  float zm = -3.0e38f;
  for (int n = 0; n < NN; ++n) {
    const float* g = &sG[n * 16];
    const float rsr = g[0] * q0 + g[1] * q1 + g[2] * q2 +
                      g[3] * q3 + g[4] * q4 + g[5] * q5;
    const float rsm = g[8] * rx + g[9] * ry + g[10] * rz;
    zm = fmaxf(zm, rsm * frcp(fmaxf(rsr, EPSF)));
  }
  red[tid] = zm;
  __syncthreads();
  for (int s = 128; s > 0; s >>= 1) {
    if (tid < s) red[tid] = fmaxf(red[tid], red[tid + s]);
    __syncthreads();
  }
  if (tid == 0) atomicMax((unsigned*)scal + (b * 4 + 2), enc_f(red[0]));
}

// ===========================================================================
// Kernel 3: render. Each wave owns 16 pixels; Gaussians tiled by 16 (M dim).
//   WMMA f32 16x16x4:  D[g,p] += A[g,k] * B[k,p]
//   A layout (16x4 f32): lanes 0-15 -> K=vgpr, lanes 16-31 -> K=vgpr+2
//   B layout (4x16 f32): vgpr v, lanes 0-15 -> K=v, lanes 16-31 -> K=v+2
//   D layout: lane = pixel (N), vgpr (+8 for upper half) = gaussian (M)
// ===========================================================================
__global__ void IRC_render_kernel(const float* __restrict__ gfeat,
                                  const float* __restrict__ scal,
                                  const float* __restrict__ Kmat,
                                  const float* __restrict__ bgapp,
                                  float* __restrict__ out) {
  const int b = blockIdx.y;
  const int tid = threadIdx.x;
  __shared__ float sG[NN * 16];
  {
    float4* d = (float4*)sG;
    const float4* s = (const float4*)(gfeat + b * NN * 16);
    for (int i = tid; i < NN * 4; i += 256) d[i] = s[i];
  }
  __syncthreads();

  const int lane = tid & 31;
  const int wid = tid >> 5;
  const int row = lane & 15;          // pixel slot / gaussian row within tile
  const bool up = lane >= 16;
  const int koff = up ? 2 : 0;

  const int P = blockIdx.x * 128 + wid * 16 + row;   // 8 waves * 16 pixels
  const int x = P & (WW - 1);
  const int y = P >> 8;
  const float fx = Kmat[b * 9 + 0], fy = Kmat[b * 9 + 4];
  const float cx = Kmat[b * 9 + 2], cy = Kmat[b * 9 + 5];
  float rx = ((float)x - cx) / fx;
  float ry = ((float)y - cy) / fy;
  float rz = 1.0f;
  const float inv = rsqrtf(rx * rx + ry * ry + 1.0f);
  rx *= inv; ry *= inv; rz *= inv;

  // B operand vgprs: pixel feature matrix, K x 16 pixels.
  v2f B0, B1, Bm;
  B0.x = up ? rz * rz : rx * rx;   // K2 : K0
  B0.y = up ? rx * ry : ry * ry;   // K3 : K1
  B1.x = up ? 0.0f : rx * rz;      // K6 : K4
  B1.y = up ? 0.0f : ry * rz;      // K7 : K5
  Bm.x = up ? rz : rx;             // K2 : K0  (pairs with p2 : p0)
  Bm.y = up ? 0.0f : ry;           // K3 : K1  (pairs with 0  : p1)

  const float alpha = scal[b * 4 + 0];
  const float acoef = scal[b * 4 + 1];
  const float inva = frcp(alpha);

  float accr = 0.f, accg = 0.f, accb = 0.f, accw = 0.f;

  for (int cg = 0; cg < NN / 16; ++cg) {
    const float* gp = &sG[(cg * 16 + row) * 16];
    const v2f A0 = *(const v2f*)(gp + 0 + koff);   // rsr features k0..1 / k2..3
    const v2f A1 = *(const v2f*)(gp + 4 + koff);   // rsr features k4..7
    const v2f Am = *(const v2f*)(gp + 8 + koff);   // rsm features p0..p2,0

    v8f Cr = {0.f, 0.f, 0.f, 0.f, 0.f, 0.f, 0.f, 0.f};
    Cr = __builtin_amdgcn_wmma_f32_16x16x4_f32(false, A0, false, B0,
                                               (short)0, Cr, false, false);
    Cr = __builtin_amdgcn_wmma_f32_16x16x4_f32(false, A1, false, B1,
                                               (short)0, Cr, false, false);
    v8f Cm = {0.f, 0.f, 0.f, 0.f, 0.f, 0.f, 0.f, 0.f};
    Cm = __builtin_amdgcn_wmma_f32_16x16x4_f32(false, Am, false, Bm,
                                               (short)0, Cm, false, false);

#pragma unroll
    for (int j = 0; j < 8; ++j) {
      const int g = cg * 16 + j + (up ? 8 : 0);
      const float rsr = Cr[j];
      const float rsm = Cm[j];
      // 1/max(rsr,eps): single v_rcp_f32
      const float rinv = frcp(fmaxf(rsr, EPSF));
      const float z = rsm * rinv;
      const float4 am = *(const float4*)&sG[g * 16 + 12];  // appR,G,B, msm
      const float d = am.w - rsm * rsm * rinv;
      const float e = fminf(__expf(fminf(-d * inva, 8.0f)), 1.0e8f);
      // 1/max(sqrt(max(rsr,0)),1e-8) == rsq(max(rsr,1e-16)): single v_rsq_f32
      const float binv = frsq(fmaxf(rsr, 1.0e-16f));
      const float integ = fmaxf(acoef * binv * e, 0.0f);
      const float z2 = z * z;
      const float w = integ * frcp(1.0f + z2 * z2);
      accr += w * am.x;
      accg += w * am.y;
      accb += w * am.z;
      accw += w;
    }
  }

  // merge the two half-wave gaussian partitions (same pixel set)
  accr += __shfl_xor(accr, 16, 32);
  accg += __shfl_xor(accg, 16, 32);
  accb += __shfl_xor(accb, 16, 32);
  accw += __shfl_xor(accw, 16, 32);

  if (lane < 16) {
    const float zbg = BETAC * dec_f(((const unsigned*)scal)[b * 4 + 2]);
    float z4 = zbg * zbg; z4 *= z4;
    const float wbg = acoef * frcp(1.0f + z4);  // erfc term clips to exactly 1
    const float dinv = frcp(fmaxf(accw + wbg, EPSF));
    const float br = bgapp[b * 3 + 0];
    const float bg = bgapp[b * 3 + 1];
    const float bbv = bgapp[b * 3 + 2];
    const int HW = HH * WW;
    out[(b * 3 + 0) * HW + P] = (accr + wbg * br) * dinv;
    out[(b * 3 + 1) * HW + P] = (accg + wbg * bg) * dinv;
    out[(b * 3 + 2) * HW + P] = (accb + wbg * bbv) * dinv;
  }
}

// ===========================================================================
extern "C" void kernel_launch(void* const* d_in, const int* in_sizes, int n_in,
                              void* d_out, int out_size, void* d_ws,
                              size_t ws_size, hipStream_t stream) {
  (void)in_sizes; (void)n_in; (void)out_size; (void)ws_size;
  const float* mu  = (const float*)d_in[0];
  const float* rho = (const float*)d_in[1];
  const float* lam = (const float*)d_in[2];
  const float* app = (const float*)d_in[3];
  const float* bga = (const float*)d_in[4];
  const float* K   = (const float*)d_in[5];
  float* gfeat = (float*)d_ws;                 // B*N*16 floats = 32 KB
  float* scal  = gfeat + BB * NN * 16;         // B*4 scalar slots
  float* out = (float*)d_out;

  IRC_prep_kernel<<<BB, NN, 0, stream>>>(mu, rho, lam, app, gfeat, scal);

  dim3 gz(HH * WW / 256, BB);
  IRC_zmax_kernel<<<gz, 256, 0, stream>>>(gfeat, K, scal);

  dim3 gr(HH * WW / 128, BB);
  IRC_render_kernel<<<gr, 256, 0, stream>>>(gfeat, scal, K, bga, out);
}